// DecodeLayer_28690381537764
// MI455X (gfx1250) — compile-verified
//
#include <hip/hip_runtime.h>
#include <stdint.h>

#define AS1 __attribute__((address_space(1)))
#define AS3 __attribute__((address_space(3)))
typedef int v4i __attribute__((ext_vector_type(4)));

// ---- CDNA5 async global->LDS copy (ASYNCcnt-tracked) ----------------------
__device__ __forceinline__ void async_g2l_b128(void* l, const void* g) {
#if defined(__AMDGCN__)
#if __has_builtin(__builtin_amdgcn_global_load_async_to_lds_b128)
  __builtin_amdgcn_global_load_async_to_lds_b128((AS1 v4i*)(void*)g,
                                                 (AS3 v4i*)l, 0, 0);
#else
  uint32_t loff = (uint32_t)(uint64_t)l;   // generic LDS ptr low 32 bits = LDS offset
  uint64_t ga   = (uint64_t)g;
  asm volatile("global_load_async_to_lds_b128 %0, %1, off"
               :: "v"(loff), "v"(ga) : "memory");
#endif
#endif
}

__device__ __forceinline__ void wait_async_zero() {
#if defined(__AMDGCN__)
#if __has_builtin(__builtin_amdgcn_s_wait_asynccnt)
  __builtin_amdgcn_s_wait_asynccnt(0);
#else
  asm volatile("s_wait_asynccnt 0" ::: "memory");
#endif
#endif
}

// ---- main decode kernel (defined FIRST so the disasm snippet shows it) ----
// logits: (B, N, 25) fp32 ; anchors: (N, 4) fp32
// out layout (floats): boxes[4*BN] | scores[BN] | classes[BN] | valid[BN] | num[B]
__global__ __launch_bounds__(256) void decode_kernel(
    const float* __restrict__ logits, const float* __restrict__ anchors,
    float* __restrict__ out, int N, long long BN) {
  __shared__ __align__(16) float tile[256 * 25];  // 25.6 KB of 320 KB/WGP
  __shared__ float wcnt[8];

  const int b   = blockIdx.y;
  const int n0  = blockIdx.x * 256;
  const int rem = N - n0;
  const int rec = rem < 256 ? rem : 256;          // 256 or 28; both % 4 == 0
  const int t   = threadIdx.x;

  // Stage rec*100 bytes of logits contiguously into LDS via async B128 packets.
  const float* gbase = logits + ((long long)b * N + n0) * 25;
  const int npkt = (rec * 25) >> 2;               // 16B packets (chunk is 16B-divisible)
  for (int p = t; p < npkt; p += 256)
    async_g2l_b128((char*)tile + ((size_t)p << 4),
                   (const char*)gbase + ((size_t)p << 4));
  wait_async_zero();       // each wave waits for its own async ops
  __syncthreads();         // then all waves see the staged tile

  const bool active = t < rec;
  int cls = 0;
  if (active) {
    const float* r = tile + t * 25;               // dword stride 25 ⊥ 64 banks: conflict-free
    const int n = n0 + t;
    float4 a = *(const float4*)(anchors + (long long)n * 4);
    float aw = a.z - a.x, ah = a.w - a.y;
    float cx = (a.x + a.z) * 0.5f, cy = (a.y + a.w) * 0.5f;
    float x  = r[0] * aw + cx;
    float y  = r[1] * ah + cy;
    float hw = expf(r[2]) * aw * 0.5f;
    float hh = expf(r[3]) * ah * 0.5f;
    float4 box;
    box.x = fminf(fmaxf(x - hw, 0.0f), 1.0f);
    box.y = fminf(fmaxf(y - hh, 0.0f), 1.0f);
    box.z = fminf(fmaxf(x + hw, 0.0f), 1.0f);
    box.w = fminf(fmaxf(y + hh, 0.0f), 1.0f);

    float m = r[4];                               // argmax over 21 classes, first-max wins
    #pragma unroll
    for (int c = 1; c < 21; ++c) {
      float v = r[4 + c];
      if (v > m) { m = v; cls = c; }
    }

    long long idx = (long long)b * N + n;
    *(float4*)(out + idx * 4) = box;              // coalesced 16B stores
    out[BN * 4 + idx] = m;                        // scores
    out[BN * 5 + idx] = (float)cls;               // classes (numeric, fp32 buffer)
    out[BN * 6 + idx] = (cls != 0) ? 1.0f : 0.0f; // valid
  }

  // Per-block count of valid detections -> one exact fp32 atomic per block.
  unsigned long long bal = __ballot(active && (cls != 0));
  if ((t & 31) == 0) wcnt[t >> 5] = (float)__popcll(bal);
  __syncthreads();
  if (t == 0) {
    float s = 0.0f;
    #pragma unroll
    for (int w = 0; w < 8; ++w) s += wcnt[w];
    atomicAdd(out + BN * 7 + b, s);               // integer-valued: exact & deterministic
  }
}

// ---- zero the num[] region (d_out is poisoned before timing) --------------
__global__ void zero_num_kernel(float* __restrict__ num, int n) {
  int i = blockIdx.x * blockDim.x + threadIdx.x;
  if (i < n) num[i] = 0.0f;
}

extern "C" void kernel_launch(void* const* d_in, const int* in_sizes, int n_in,
                              void* d_out, int out_size, void* d_ws, size_t ws_size,
                              hipStream_t stream) {
  const float* logits  = (const float*)d_in[0];
  const float* anchors = (const float*)d_in[1];
  float* out = (float*)d_out;

  const int N  = in_sizes[1] / 4;                 // 8732
  const int B  = (int)(in_sizes[0] / ((long long)N * 25)); // 128
  const long long BN = (long long)B * N;

  zero_num_kernel<<<(B + 127) / 128, 128, 0, stream>>>(out + BN * 7, B);

  dim3 grid((N + 255) / 256, B);
  decode_kernel<<<grid, 256, 0, stream>>>(logits, anchors, out, N, BN);
}